// SampleAndGroup_31413390803130
// MI455X (gfx1250) — compile-verified
//
#include <hip/hip_runtime.h>

#define RADIUS2   0.04f
#define NSAMPLE_  32
#define B_        8
#define N_        16384
#define NPOINT_   2048
#define C_        64

typedef __attribute__((ext_vector_type(2))) float v2f;
typedef __attribute__((ext_vector_type(8))) float v8f;

__device__ __forceinline__ unsigned ballot32(bool p) {
#if __has_builtin(__builtin_amdgcn_ballot_w32)
    return __builtin_amdgcn_ballot_w32(p);
#else
    return (unsigned)__ballot(p);
#endif
}

// Pack xyz points as (-2x, -2y, -2z, |x|^2) so one WMMA_F32_16X16X4 computes
// full squared distances:  D = A(px,py,pz,1) * B(-2x,-2y,-2z,|x|^2) + C(|p|^2)
__global__ void prep_pts_kernel(const float* __restrict__ xyz,
                                float4* __restrict__ pts4, int total) {
    int i = blockIdx.x * blockDim.x + threadIdx.x;
    if (i < total) {
        float x = xyz[3 * i + 0];
        float y = xyz[3 * i + 1];
        float z = xyz[3 * i + 2];
        pts4[i] = make_float4(-2.0f * x, -2.0f * y, -2.0f * z,
                              x * x + y * y + z * z);
    }
}

// ---------------------------------------------------------------------------
// Scan kernel: WMMA distance tiles + ordered first-32 selection, writes the
// finalized (B,NPOINT,NSAMPLE) int32 index lists to global scratch.
// One wave owns a 16-query tile. Counts live in registers (ballot is wave-
// uniform), so the hot loop is WMMA + VALU + at most 16 predicated ds_stores.
// ---------------------------------------------------------------------------
__global__ __launch_bounds__(128) void ballquery_kernel(
    const float*  __restrict__ new_xyz,  // (B,NPOINT,3)
    const float4* __restrict__ pts4,     // (B,N) packed
    int*          __restrict__ idxbuf)   // (B,NPOINT,NSAMPLE)
{
    __shared__ int   s_list[4][16][NSAMPLE_ + 1];  // +1 trash slot
    __shared__ int   s_cnt[4][16];
    __shared__ float s_q[4][16][4];                // qx,qy,qz,|q|^2

    const int wave = threadIdx.x >> 5;
    const int lane = threadIdx.x & 31;
    const int hh   = lane >> 4;       // half: 0 = lanes 0-15, 1 = lanes 16-31
    const int lp   = lane & 15;

    const int blocksPerBatch = (NPOINT_ / 16) / 4;      // 32
    const int b    = blockIdx.x / blocksPerBatch;
    const int tile = (blockIdx.x % blocksPerBatch) * 4 + wave;
    const int p0   = tile * 16;

    // ---- load this wave's 16 query points (row = lp, duplicated per half) ----
    const float* q = new_xyz + ((size_t)b * NPOINT_ + p0) * 3;
    float qx = q[lp * 3 + 0];
    float qy = q[lp * 3 + 1];
    float qz = q[lp * 3 + 2];
    float qn = qx * qx + qy * qy + qz * qz;
    if (lane < 16) {
        s_q[wave][lp][0] = qx; s_q[wave][lp][1] = qy;
        s_q[wave][lp][2] = qz; s_q[wave][lp][3] = qn;
    }
    __syncthreads();

    // A operand, 16x4 f32 layout: lanes 0-15 hold (K0,K1), lanes 16-31 (K2,K3)
    v2f A;
    A.x = hh ? qz   : qx;
    A.y = hh ? 1.0f : qy;

    // C operand: C[j] = |q_row|^2, row = j + 8*half (C/D VGPR layout)
    v8f C;
#pragma unroll
    for (int j = 0; j < 8; ++j) C[j] = s_q[wave][j + 8 * hh][3];

    // per-lane (wave-uniform within each half) hit counts for 8 rows
    int cnts[8];
#pragma unroll
    for (int j = 0; j < 8; ++j) cnts[j] = 0;

    const unsigned below = (1u << lp) - 1u;
    const float4* pb = pts4 + (size_t)b * N_;

    // ---- scan 32 points / iteration: two WMMAs, register-only bookkeeping ----
    for (int it = 0; it < N_ / 32; ++it) {
        const int n0 = it * 32;

        const float* p0p = (const float*)(pb + n0 + lp);
        const float* p1p = (const float*)(pb + n0 + 16 + lp);
        v2f B0, B1;
        B0.x = p0p[hh * 2 + 0]; B0.y = p0p[hh * 2 + 1];
        B1.x = p1p[hh * 2 + 0]; B1.y = p1p[hh * 2 + 1];

        if (n0 + 512 < N_)  // stream ahead into WGP cache
            __builtin_prefetch((const char*)(pb + n0 + 512), 0, 3);

        v8f D0, D1;
#if __has_builtin(__builtin_amdgcn_wmma_f32_16x16x4_f32)
        D0 = __builtin_amdgcn_wmma_f32_16x16x4_f32(false, A, false, B0,
                                                   (short)0, C, false, false);
        D1 = __builtin_amdgcn_wmma_f32_16x16x4_f32(false, A, false, B1,
                                                   (short)0, C, false, false);
#else
        {
            float P0x = p0p[0] , P0y = p0p[1], P0z = p0p[2], P0w = p0p[3];
            float P1x = p1p[0] , P1y = p1p[1], P1z = p1p[2], P1w = p1p[3];
#pragma unroll
            for (int j = 0; j < 8; ++j) {
                int row = j + 8 * hh;
                float rx = s_q[wave][row][0], ry = s_q[wave][row][1],
                      rz = s_q[wave][row][2];
                D0[j] = C[j] + P0w + P0x * rx + P0y * ry + P0z * rz;
                D1[j] = C[j] + P1w + P1x * rx + P1y * ry + P1z * rz;
            }
        }
#endif

#pragma unroll
        for (int j = 0; j < 8; ++j) {
            unsigned m   = ballot32(D0[j] <= RADIUS2);        // wave-uniform
            unsigned sub = (m >> (16 * hh)) & 0xFFFFu;
            int slot = cnts[j] + __builtin_popcount(sub & below);
            if ((sub >> lp) & 1u)
                s_list[wave][j + 8 * hh][slot < NSAMPLE_ ? slot : NSAMPLE_] =
                    n0 + lp;
            cnts[j] += __builtin_popcount(sub);
        }
#pragma unroll
        for (int j = 0; j < 8; ++j) {
            unsigned m   = ballot32(D1[j] <= RADIUS2);
            unsigned sub = (m >> (16 * hh)) & 0xFFFFu;
            int slot = cnts[j] + __builtin_popcount(sub & below);
            if ((sub >> lp) & 1u)
                s_list[wave][j + 8 * hh][slot < NSAMPLE_ ? slot : NSAMPLE_] =
                    n0 + 16 + lp;
            cnts[j] += __builtin_popcount(sub);
        }

        // early exit once every row has 32 hits (expected after ~6% of N)
        if ((it & 3) == 3) {
            int mn = cnts[0];
#pragma unroll
            for (int j = 1; j < 8; ++j) mn = mn < cnts[j] ? mn : cnts[j];
            if (ballot32(mn >= NSAMPLE_) == 0xFFFFFFFFu) break;
        }
    }

    // publish counts to LDS (one lane per half), then finalize + store lists
    if (lp == 0) {
#pragma unroll
        for (int j = 0; j < 8; ++j) s_cnt[wave][j + 8 * hh] = cnts[j];
    }
    for (int m = 0; m < 16; ++m) {
        int cnt   = s_cnt[wave][m];
        int first = (cnt > 0) ? s_list[wave][m][0] : (N_ - 1);
        int idx   = (lane < cnt) ? s_list[wave][m][lane] : first;
        idxbuf[((size_t)b * NPOINT_ + (p0 + m)) * NSAMPLE_ + lane] = idx;
    }
}

// ---------------------------------------------------------------------------
// Gather kernel: pure bandwidth phase. One wave per (b, p, s) sample row:
// coalesced 256B feature-row read, coalesced 268B output segment write.
// ---------------------------------------------------------------------------
__global__ __launch_bounds__(256) void gather_kernel(
    const float* __restrict__ xyz,      // (B,N,3)
    const float* __restrict__ new_xyz,  // (B,NPOINT,3)
    const float* __restrict__ feat,     // (B,N,C)
    const int*   __restrict__ idxbuf,   // (B,NPOINT,NSAMPLE)
    float*       __restrict__ out)      // (B,NPOINT,NSAMPLE,67)
{
    const int wave = threadIdx.x >> 5;
    const int lane = threadIdx.x & 31;
    const size_t t = (size_t)blockIdx.x * 8 + wave;   // (b*NPOINT+p)*32 + s

    const int    idx = idxbuf[t];
    const size_t bp  = t >> 5;                        // b*NPOINT + p
    const int    b   = (int)(bp / NPOINT_);

    const float* fr = feat + ((size_t)b * N_ + idx) * C_;
    float*       o  = out + t * 67;

    if (lane < 3)
        o[lane] = xyz[((size_t)b * N_ + idx) * 3 + lane]
                - new_xyz[bp * 3 + lane];
    o[3 + lane]  = fr[lane];
    o[35 + lane] = fr[lane + 32];
}

extern "C" void kernel_launch(void* const* d_in, const int* in_sizes, int n_in,
                              void* d_out, int out_size, void* d_ws, size_t ws_size,
                              hipStream_t stream) {
    const float* xyz     = (const float*)d_in[0];
    const float* new_xyz = (const float*)d_in[1];
    const float* feat    = (const float*)d_in[2];

    float4* pts4   = (float4*)d_ws;                        // 2 MB
    int*    idxbuf = (int*)((char*)d_ws + (size_t)B_ * N_ * sizeof(float4));
    float*  out    = (float*)d_out;

    const int total = B_ * N_;
    prep_pts_kernel<<<(total + 255) / 256, 256, 0, stream>>>(xyz, pts4, total);

    const int scan_blocks = B_ * ((NPOINT_ / 16) / 4);     // 256 x 128 threads
    ballquery_kernel<<<scan_blocks, 128, 0, stream>>>(new_xyz, pts4, idxbuf);

    const int gather_blocks = (B_ * NPOINT_ * NSAMPLE_) / 8;  // 65536 x 256 thr
    gather_kernel<<<gather_blocks, 256, 0, stream>>>(xyz, new_xyz, feat,
                                                     idxbuf, out);
}